// ChannelWiseQuantumKernel_70978629533756
// MI455X (gfx1250) — compile-verified
//
#include <hip/hip_runtime.h>

#define IN_CH 4
#define KSZ 3
#define NUM_KERNELS 2
#define NPOS 9
#define OUT_HW 62
#define PPI (OUT_HW * OUT_HW)   // 3844 patches per image
#define DIM 16                  // 2^4 amplitudes
#define TOTAL_W 108

typedef __attribute__((ext_vector_type(2))) float v2f;
typedef __attribute__((ext_vector_type(8))) float v8f;

// ------------------------------------------------------------------
// Phase 1: build the 18 per-(kernel,pos) 16x16 complex weight unitaries.
// Thread j simulates the weight sub-circuit starting from basis |j> and
// writes column j of U. Qubit q lives on bit (3-q) of the amplitude index.
// ------------------------------------------------------------------
__global__ void build_unitaries(const float* __restrict__ weights,
                                float* __restrict__ Ur,
                                float* __restrict__ Ui) {
  const int id   = blockIdx.x;        // kidx*9 + pos
  const int kidx = id / NPOS;
  const int pos  = id % NPOS;
  const int j    = threadIdx.x;
  if (j >= DIM) return;

  float ar[DIM], ai[DIM];
#pragma unroll
  for (int m = 0; m < DIM; ++m) { ar[m] = (m == j) ? 1.0f : 0.0f; ai[m] = 0.0f; }

  const float* w = weights + kidx * TOTAL_W + pos * (IN_CH * 3);

#pragma unroll
  for (int q = 0; q < IN_CH; ++q) {
    const int mask = 8 >> q;          // qubit q -> bit (3-q)
    float th, c, s;
    // RX(th): [[c, -i s],[-i s, c]]
    th = w[q * 3 + 0]; c = cosf(0.5f * th); s = sinf(0.5f * th);
#pragma unroll
    for (int m = 0; m < DIM; ++m) if (!(m & mask)) {
      const int m1 = m | mask;
      float a0r = ar[m], a0i = ai[m], a1r = ar[m1], a1i = ai[m1];
      ar[m]  = c * a0r + s * a1i;  ai[m]  = c * a0i - s * a1r;
      ar[m1] = c * a1r + s * a0i;  ai[m1] = c * a1i - s * a0r;
    }
    // RY(th): [[c, -s],[s, c]]
    th = w[q * 3 + 1]; c = cosf(0.5f * th); s = sinf(0.5f * th);
#pragma unroll
    for (int m = 0; m < DIM; ++m) if (!(m & mask)) {
      const int m1 = m | mask;
      float a0r = ar[m], a0i = ai[m], a1r = ar[m1], a1i = ai[m1];
      ar[m]  = c * a0r - s * a1r;  ai[m]  = c * a0i - s * a1i;
      ar[m1] = s * a0r + c * a1r;  ai[m1] = s * a0i + c * a1i;
    }
    // RZ(th): diag(e^{-i th/2}, e^{+i th/2})
    th = w[q * 3 + 2]; c = cosf(0.5f * th); s = sinf(0.5f * th);
#pragma unroll
    for (int m = 0; m < DIM; ++m) {
      const float r0 = ar[m], i0 = ai[m];
      if (m & mask) { ar[m] = c * r0 - s * i0; ai[m] = c * i0 + s * r0; }
      else          { ar[m] = c * r0 + s * i0; ai[m] = c * i0 - s * r0; }
    }
  }
  // CX ring (0,1)(1,2)(2,3)(3,0): bits (8,4)(4,2)(2,1)(1,8)
  const int cb[4] = {8, 4, 2, 1};
  const int tb[4] = {4, 2, 1, 8};
#pragma unroll
  for (int e = 0; e < 4; ++e) {
#pragma unroll
    for (int m = 0; m < DIM; ++m) if ((m & cb[e]) && !(m & tb[e])) {
      const int m1 = m | tb[e];
      float t;
      t = ar[m]; ar[m] = ar[m1]; ar[m1] = t;
      t = ai[m]; ai[m] = ai[m1]; ai[m1] = t;
    }
  }
  float* ur = Ur + id * DIM * DIM;
  float* ui = Ui + id * DIM * DIM;
#pragma unroll
  for (int m = 0; m < DIM; ++m) { ur[m * DIM + j] = ar[m]; ui[m * DIM + j] = ai[m]; }
}

// ------------------------------------------------------------------
// Phase 2: one wave (32 lanes) carries 16 patches' quantum states in the
// WMMA f32 C/D layout: lane L holds rows r+8*(L>>4), patch column L&15.
// Per position: data RY gates in registers, then newS = U * S via
// 16 x v_wmma_f32_16x16x4_f32 (complex as 4 real 16x16x16 matmuls).
// ------------------------------------------------------------------
__global__ void __launch_bounds__(32)
qconv_wmma(const float* __restrict__ x,
           const float* __restrict__ Ur,
           const float* __restrict__ Ui,
           float* __restrict__ out) {
  const int lane = threadIdx.x;
  const int half = lane >> 4;
  const int pcol = lane & 15;
  const int kidx = blockIdx.y;
  const int P    = blockIdx.x * 16 + pcol;   // global patch id
  const int b    = P / PPI;
  const int rem  = P % PPI;
  const int oh   = rem / OUT_HW;
  const int ow   = rem % OUT_HW;
  const float* xb = x + (size_t)b * IN_CH * 64 * 64;

  __shared__ float lr[DIM][DIM];
  __shared__ float li[DIM][DIM];

  float sr[8], si[8];                         // state, C/D layout
#pragma unroll
  for (int r = 0; r < 8; ++r) { sr[r] = 0.25f; si[r] = 0.0f; }  // |+...+>

  for (int pos = 0; pos < NPOS; ++pos) {
    const int kh = pos / KSZ, kw = pos % KSZ;

    // ---- data RY gates (qubit = channel) ----
#pragma unroll
    for (int ch = 0; ch < IN_CH; ++ch) {
      const float th = xb[(ch * 64 + (oh + kh)) * 64 + (ow + kw)];
      const float c = cosf(0.5f * th), s = sinf(0.5f * th);
      if (ch == 0) {
        // bit3: rows m <-> m+8 live in opposite wave halves
#pragma unroll
        for (int r = 0; r < 8; ++r) {
          const float tr = __shfl_xor(sr[r], 16, 32);
          const float ti = __shfl_xor(si[r], 16, 32);
          const float lo_r = c * sr[r] - s * tr, hi_r = s * tr + c * sr[r];
          const float lo_i = c * si[r] - s * ti, hi_i = s * ti + c * si[r];
          sr[r] = half ? hi_r : lo_r;
          si[r] = half ? hi_i : lo_i;
        }
      } else {
        const int m_ = 8 >> ch;               // 4,2,1 : in-lane pairs
#pragma unroll
        for (int r = 0; r < 8; ++r) if (!(r & m_)) {
          const int r1 = r | m_;
          float a0r = sr[r], a0i = si[r], a1r = sr[r1], a1i = si[r1];
          sr[r]  = c * a0r - s * a1r;  si[r]  = c * a0i - s * a1i;
          sr[r1] = s * a0r + c * a1r;  si[r1] = s * a0i + c * a1i;
        }
      }
    }

    // ---- stage state into LDS as [row][patch] for the B operand ----
    __syncthreads();
#pragma unroll
    for (int r = 0; r < 8; ++r) {
      const int row = r + 8 * half;
      lr[row][pcol] = sr[r];
      li[row][pcol] = si[r];
    }
    __syncthreads();

    // ---- complex 16x16x16 matmul: D = U * S via 4 K-chunks ----
    const float* urp = Ur + (kidx * NPOS + pos) * 256;
    const float* uip = Ui + (kidx * NPOS + pos) * 256;
    v8f accR = {};
    v8f accI = {};
#pragma unroll
    for (int cch = 0; cch < 4; ++cch) {
      const int k0 = 4 * cch + 2 * half;      // A layout: K = 4c + v + 2*half
      v2f aR, aI, aIn, bR, bI;
      aR.x = urp[pcol * 16 + k0]; aR.y = urp[pcol * 16 + k0 + 1];
      aI.x = uip[pcol * 16 + k0]; aI.y = uip[pcol * 16 + k0 + 1];
      aIn = -aI;
      const int rb = 4 * cch + 2 * half;      // B layout: row = 4c + v + 2*half
      bR.x = lr[rb][pcol]; bR.y = lr[rb + 1][pcol];
      bI.x = li[rb][pcol]; bI.y = li[rb + 1][pcol];
      // Re(D) += Ur*Sr - Ui*Si ; Im(D) += Ur*Si + Ui*Sr
      accR = __builtin_amdgcn_wmma_f32_16x16x4_f32(false, aR,  false, bR, (short)0, accR, false, false);
      accR = __builtin_amdgcn_wmma_f32_16x16x4_f32(false, aIn, false, bI, (short)0, accR, false, false);
      accI = __builtin_amdgcn_wmma_f32_16x16x4_f32(false, aR,  false, bI, (short)0, accI, false, false);
      accI = __builtin_amdgcn_wmma_f32_16x16x4_f32(false, aI,  false, bR, (short)0, accI, false, false);
    }
#pragma unroll
    for (int r = 0; r < 8; ++r) { sr[r] = accR[r]; si[r] = accI[r]; }
  }

  // ---- <Z_q> = sum_m (-1)^{bit(3-q) of m} |amp_m|^2 ----
  float z[4] = {0.f, 0.f, 0.f, 0.f};
#pragma unroll
  for (int r = 0; r < 8; ++r) {
    const float p = sr[r] * sr[r] + si[r] * si[r];
    z[0] += half    ? -p : p;   // qubit0 -> bit3 (wave half)
    z[1] += (r & 4) ? -p : p;   // qubit1 -> bit2
    z[2] += (r & 2) ? -p : p;   // qubit2 -> bit1
    z[3] += (r & 1) ? -p : p;   // qubit3 -> bit0
  }
#pragma unroll
  for (int q = 0; q < 4; ++q) z[q] += __shfl_xor(z[q], 16, 32);

  if (lane < 16) {
    float* o = out + ((size_t)(b * (NUM_KERNELS * IN_CH) + kidx * IN_CH) * OUT_HW + oh) * OUT_HW + ow;
#pragma unroll
    for (int q = 0; q < 4; ++q) o[(size_t)q * PPI] = z[q];
  }
}

extern "C" void kernel_launch(void* const* d_in, const int* in_sizes, int n_in,
                              void* d_out, int out_size, void* d_ws, size_t ws_size,
                              hipStream_t stream) {
  const float* x       = (const float*)d_in[0];   // (16,4,64,64)
  const float* weights = (const float*)d_in[1];   // (2,108)
  float* Ur = (float*)d_ws;                                   // 18*256 floats
  float* Ui = Ur + NUM_KERNELS * NPOS * DIM * DIM;            // 18*256 floats

  build_unitaries<<<NUM_KERNELS * NPOS, 32, 0, stream>>>(weights, Ur, Ui);

  dim3 grid(PPI /* 3844 waves * 16 patches = 61504 patches */, NUM_KERNELS);
  qconv_wmma<<<grid, 32, 0, stream>>>(x, Ur, Ui, (float*)d_out);
}